// Gradient_loss_52450140619130
// MI455X (gfx1250) — compile-verified
//
#include <hip/hip_runtime.h>

// ---------------------------------------------------------------------------
// Sobel gradient-magnitude mean on [64,1,512,512] fp32, SAME zero padding.
// Memory-bound (64 MiB in, 1 float out):
//   - float4 (global_load_b128) streaming, 4 columns/thread
//   - separable stencil, 3-row register ring buffer
//   - global_prefetch_b8 read-ahead
//   - V_WMMA_F32_16X16X4_F32 (exact fp32) for the 256->1 block reduction
//   - global_atomic_add_f32 for the grid combine
// ---------------------------------------------------------------------------

typedef float v2f __attribute__((ext_vector_type(2)));
typedef float v8f __attribute__((ext_vector_type(8)));

#define IMG_H 512
#define IMG_W 512
#define N_IMG 64
#define ROWS_PER_TY 32          // blockDim.y = 2 -> 64 rows per block
#define TX 128                  // 128 col-threads * 4 cols = 512 (full width)

// 0.1 * (1/N) folded into the per-block partial before the atomic add.
#define OUT_SCALE (0.1f / (64.0f * 512.0f * 512.0f))

__global__ void gl_zero_out(float* out) { out[0] = 0.0f; }

// Horizontal pass for 4 consecutive columns of one row:
//   th[i] = x[i-1] + 2x[i] + x[i+1],  tv[i] = x[i-1] - x[i+1]
// Zero padding outside the image (rows and columns).
__device__ __forceinline__ void hpass4(const float* __restrict__ img, int y,
                                       int c0, float th[4], float tv[4]) {
  if (y < 0 || y >= IMG_H) {
#pragma unroll
    for (int i = 0; i < 4; ++i) { th[i] = 0.0f; tv[i] = 0.0f; }
    return;
  }
  const float* row = img + (size_t)y * IMG_W;
  float4 v = *(const float4*)(row + c0);                  // global_load_b128
  float a0 = (c0 > 0)            ? row[c0 - 1] : 0.0f;    // cached halo
  float a5 = (c0 + 4 < IMG_W)    ? row[c0 + 4] : 0.0f;    // cached halo
  float a[6] = {a0, v.x, v.y, v.z, v.w, a5};
#pragma unroll
  for (int i = 0; i < 4; ++i) {
    th[i] = a[i] + 2.0f * a[i + 1] + a[i + 2];
    tv[i] = a[i] - a[i + 2];
  }
}

__global__ void __launch_bounds__(256)
gl_sobel_mean_kernel(const float* __restrict__ x, float* __restrict__ out) {
  __shared__ float smem[256];

  const int img_idx = blockIdx.y;                 // 0..63
  const int band    = blockIdx.x;                 // 8 row bands of 64
  const float* img  = x + (size_t)img_idx * IMG_H * IMG_W;

  const int c0 = 4 * (int)threadIdx.x;                        // 0..508
  const int r0 = band * 2 * ROWS_PER_TY + (int)threadIdx.y * ROWS_PER_TY;

  // 3-row register ring buffer of horizontal-pass results, x4 columns.
  float thm[4], tvm[4], th0[4], tv0[4], thp[4], tvp[4];
  hpass4(img, r0 - 1, c0, thm, tvm);
  hpass4(img, r0,     c0, th0, tv0);

  float acc = 0.0f;
  for (int y = r0; y < r0 + ROWS_PER_TY; ++y) {
    // Read-ahead for the streaming row (lowers to global_prefetch_b8).
    if (y + 2 < IMG_H)
      __builtin_prefetch(img + (size_t)(y + 2) * IMG_W + c0, 0, 1);

    hpass4(img, y + 1, c0, thp, tvp);
#pragma unroll
    for (int i = 0; i < 4; ++i) {
      // Vertical pass: gh = [1,0,-1]^T on th; gv = [1,2,1]^T on tv.
      float gh = thm[i] - thp[i];
      float gv = tvm[i] + 2.0f * tv0[i] + tvp[i];
      acc += sqrtf(gh * gh + gv * gv);
      thm[i] = th0[i]; tvm[i] = tv0[i]; th0[i] = thp[i]; tv0[i] = tvp[i];
    }
  }

  const int tid = (int)threadIdx.y * TX + (int)threadIdx.x;   // 0..255
  smem[tid] = acc;
  __syncthreads();

  // Block reduction (256 -> 1) on the fp32 matrix pipe:
  // D = A(16x4, ones) x B(4x16, partials) + C ; chaining C over 4 chunks
  // accumulates all 256 values into each row of D (rows identical).
  // Executed by wave 0 only, all 32 lanes active => EXEC all-1s (required).
  if (tid < 32) {
    const int lane = tid;
    v2f a; a.x = 1.0f; a.y = 1.0f;                 // A = ones(16x4)
    // Branchless B(4x16) f32 layout: lanes 0-15 hold {K0,K1}, lanes 16-31
    // hold {K2,K3}; element (K,N) lives at smem[base + K*16 + N].
    const int baddr = ((lane & 16) << 1) + (lane & 15);
    v8f cacc = {};
#pragma unroll
    for (int ch = 0; ch < 4; ++ch) {
      const int base = ch * 64 + baddr;
      v2f b;
      b.x = smem[base];                            // one ds_load_2addr_b32
      b.y = smem[base + 16];
      cacc = __builtin_amdgcn_wmma_f32_16x16x4_f32(
          /*neg_a=*/false, a, /*neg_b=*/false, b,
          /*c_mod=*/(short)0, cacc, /*reuse_a=*/false, /*reuse_b=*/false);
    }
    // cacc[0] on lanes 0..15 = column sums for M=0; reduce the 16 lanes.
    float v = cacc[0];
    v += __shfl_xor(v, 1, 32);
    v += __shfl_xor(v, 2, 32);
    v += __shfl_xor(v, 4, 32);
    v += __shfl_xor(v, 8, 32);
    if (lane == 0)
      atomicAdd(out, v * OUT_SCALE);               // global_atomic_add_f32
  }
}

extern "C" void kernel_launch(void* const* d_in, const int* in_sizes, int n_in,
                              void* d_out, int out_size, void* d_ws, size_t ws_size,
                              hipStream_t stream) {
  (void)in_sizes; (void)n_in; (void)out_size; (void)d_ws; (void)ws_size;
  const float* x = (const float*)d_in[0];
  float* out = (float*)d_out;

  // d_out is poisoned by the harness and never re-zeroed between replays:
  // zero it every call (deterministic, capture-safe).
  gl_zero_out<<<1, 1, 0, stream>>>(out);

  dim3 grid(IMG_H / (2 * ROWS_PER_TY), N_IMG);    // (8, 64) = 512 blocks
  dim3 block(TX, 2);                              // 256 threads = 8 waves
  gl_sobel_mean_kernel<<<grid, block, 0, stream>>>(x, out);
}